// TorchEAFTCrossEntropy_16758962389544
// MI455X (gfx1250) — compile-verified
//
#include <hip/hip_runtime.h>
#include <hip/hip_bf16.h>

#define IGNORE_INDEX (-100)
#define TOPK 20
#define BLOCK 256
#define TILE 1024  // floats per tile = BLOCK * 4

typedef __attribute__((address_space(3))) float* lds_fp;

// byte offset of a __shared__ object within the wave's LDS allocation
__device__ __forceinline__ unsigned lds_byte_off(float* p) {
    return (unsigned)(unsigned long long)(lds_fp)p;
}

// GVS-mode async global->LDS copy, 16B per lane, ASYNCcnt-tracked:
//   copies the 16B at global saddr + voff into LDS at vdst
#define ASYNC_LD_B128(LDSA, VOFF, SBASE)                                       \
    asm volatile("global_load_async_to_lds_b128 %0, %1, %2"                    \
                 :: "v"(LDSA), "v"(VOFF), "s"(SBASE) : "memory")

#define WAIT_ASYNC(IMM) asm volatile("s_wait_asynccnt " #IMM ::: "memory")

__global__ __launch_bounds__(BLOCK) void eaft_ce_row_kernel(
    const float* __restrict__ src, const int* __restrict__ tgts, int V,
    float* __restrict__ ws_num, float* __restrict__ ws_cnt) {
    __shared__ float abuf[2 * TILE];      // async staging, double buffered (8 KB)
    __shared__ float tk[BLOCK * TOPK];    // per-thread top-k lists (20 KB)
    __shared__ float red[BLOCK];          // reduction scratch
    __shared__ int   head[BLOCK];         // tournament heads (wave 0 only)
    __shared__ float topv[TOPK];          // merged global top-k
    __shared__ float tgt_sh;              // captured target logit

    const int tid = threadIdx.x;
    const int row_id = blockIdx.x;
    const float* row = src + (size_t)row_id * (size_t)V;
    const unsigned long long base = (unsigned long long)row;
    const int tgt = tgts[row_id];

    if (tid == 0) tgt_sh = 0.0f;
    __syncthreads();

    float m = -__builtin_inff();  // running max
    float s = 0.0f;               // running sum of exp(x - m)
    float tr[TOPK];
#pragma unroll
    for (int k = 0; k < TOPK; ++k) tr[k] = -__builtin_inff();

    // rows are only 4B aligned: peel 0..3 head floats so the body is 16B aligned
    const int head_n = (int)(((16u - (unsigned)(base & 15ull)) & 15u) >> 2);
    const int full = (V - head_n) / TILE;   // full 4KB tiles in the aligned body
    const int tail0 = head_n + full * TILE;

    auto issue_tile = [&](int t, int bufsel) {
        unsigned ldsa = lds_byte_off(&abuf[bufsel * TILE + tid * 4]);
        unsigned goff = (unsigned)(head_n * 4 + t * (TILE * 4) + tid * 16);
        ASYNC_LD_B128(ldsa, goff, base);  // one b128 per lane per tile
    };

    auto update = [&](float x, int idx) {
        // online logsumexp
        if (x > m) { s = s * __expf(m - x) + 1.0f; m = x; }
        else       { s += __expf(x - m); }
        // top-k maintenance (rarely-taken unrolled bubble insert)
        if (x > tr[TOPK - 1]) {
            float c = x;
#pragma unroll
            for (int k = 0; k < TOPK; ++k) {
                float hi = fmaxf(tr[k], c);
                c = fminf(tr[k], c);
                tr[k] = hi;
            }
        }
        if (idx == tgt) tgt_sh = x;
    };

    if (full >= 1) issue_tile(0, 0);
    if (full >= 2) issue_tile(1, 1);

    // head elements overlap with the in-flight async fill
    for (int idx = tid; idx < head_n; idx += BLOCK) update(row[idx], idx);

    for (int t = 0; t < full; ++t) {
        // in-order async completion: <=1 outstanding retires tile t
        if (t + 1 < full) { WAIT_ASYNC(0x1); } else { WAIT_ASYNC(0x0); }
        const int b = t & 1;
#pragma unroll
        for (int j = 0; j < 4; ++j) {  // lane owns 4 contiguous floats (ds_load_b128)
            float x = abuf[b * TILE + tid * 4 + j];
            int idx = head_n + t * TILE + tid * 4 + j;
            update(x, idx);
        }
        if (t + 2 < full) issue_tile(t + 2, b);
    }

    // ragged tail
    for (int idx = tail0 + tid; idx < V; idx += BLOCK) update(row[idx], idx);

    // ---- block logsumexp (fixed-order trees => deterministic) ----
    __syncthreads();
    red[tid] = m;
    __syncthreads();
    for (int o = BLOCK / 2; o > 0; o >>= 1) {
        if (tid < o) red[tid] = fmaxf(red[tid], red[tid + o]);
        __syncthreads();
    }
    const float M = red[0];
    __syncthreads();
    red[tid] = s * __expf(m - M);
    __syncthreads();
    for (int o = BLOCK / 2; o > 0; o >>= 1) {
        if (tid < o) red[tid] += red[tid + o];
        __syncthreads();
    }
    const float lse = M + __logf(red[0]);

    // ---- publish per-thread top-k, merge with a wave-0 tournament ----
#pragma unroll
    for (int k = 0; k < TOPK; ++k) tk[tid * TOPK + k] = tr[k];
    __syncthreads();

    if (tid < 32) {
        for (int i = tid; i < BLOCK; i += 32) head[i] = 0;
        for (int r = 0; r < TOPK; ++r) {
            float best = -__builtin_inff();
            int bi = 0;
            for (int i = tid; i < BLOCK; i += 32) {  // 8 list heads per lane
                int h = head[i];
                float v = (h < TOPK) ? tk[i * TOPK + h] : -__builtin_inff();
                if (v > best) { best = v; bi = i; }
            }
#pragma unroll
            for (int off = 16; off > 0; off >>= 1) {  // wave32 arg-max
                float ov = __shfl_xor(best, off, 32);
                int oi = __shfl_xor(bi, off, 32);
                if (ov > best) { best = ov; bi = oi; }
            }
            if (tid == 0) { topv[r] = best; head[bi] += 1; }
        }
    }
    __syncthreads();

    if (tid == 0) {
        const bool valid = (tgt != IGNORE_INDEX);
        // entropy of softmax over merged top-k
        float mx = topv[0];
        float se = 0.0f;
#pragma unroll
        for (int r = 0; r < TOPK; ++r) se += __expf(topv[r] - mx);
        float lsek = mx + __logf(se);
        float H = 0.0f;
#pragma unroll
        for (int r = 0; r < TOPK; ++r) {
            float lp = topv[r] - lsek;
            H -= __expf(lp) * lp;
        }
        float w = H * (1.0f / 3.0f);  // pow(H/3, ALPHA=1)
        ws_num[row_id] = valid ? (lse - tgt_sh) * w : 0.0f;
        ws_cnt[row_id] = valid ? 1.0f : 0.0f;
    }
}

__global__ __launch_bounds__(BLOCK) void eaft_ce_finalize(
    const float* __restrict__ ws_num, const float* __restrict__ ws_cnt, int N,
    float* __restrict__ out) {
    __shared__ float sa[BLOCK];
    __shared__ float sb[BLOCK];
    float a = 0.0f, b = 0.0f;
    for (int i = threadIdx.x; i < N; i += BLOCK) {
        a += ws_num[i];
        b += ws_cnt[i];
    }
    sa[threadIdx.x] = a;
    sb[threadIdx.x] = b;
    __syncthreads();
    for (int o = BLOCK / 2; o > 0; o >>= 1) {
        if (threadIdx.x < o) {
            sa[threadIdx.x] += sa[threadIdx.x + o];
            sb[threadIdx.x] += sb[threadIdx.x + o];
        }
        __syncthreads();
    }
    if (threadIdx.x == 0) out[0] = sa[0] / sb[0];
}

extern "C" void kernel_launch(void* const* d_in, const int* in_sizes, int n_in,
                              void* d_out, int out_size, void* d_ws, size_t ws_size,
                              hipStream_t stream) {
    const float* src = (const float*)d_in[0];
    const int* tgts = (const int*)d_in[1];
    const int N = in_sizes[1];
    const int V = in_sizes[0] / N;

    float* ws_num = (float*)d_ws;
    float* ws_cnt = ws_num + N;

    eaft_ce_row_kernel<<<N, BLOCK, 0, stream>>>(src, tgts, V, ws_num, ws_cnt);
    eaft_ce_finalize<<<1, BLOCK, 0, stream>>>(ws_num, ws_cnt, N, (float*)d_out);
}